// PowerIterationPageRank_8297876816012
// MI455X (gfx1250) — compile-verified
//
#include <hip/hip_runtime.h>
#include <hip/hip_bf16.h>

// APPNP power iteration for MI455X (gfx1250).
//
// Strategy (see analysis): build CSR once per call (atomic histogram + single-
// block scan + atomic scatter), then 5x { gather-SpMM (no atomics, L2-resident
// 20MB logits) ; WMMA combine l' = alpha*init + (1-alpha)*agg } with the dense
// combine expressed as D = ((1-a)*I16) x T + (a*init) on V_WMMA_F32_16X16X4_F32.

#define N_NODES 100000
#define N_EDGES 1600000
#define NC      50          // classes
#define ALPHA   0.15f
#define OMA     0.85f       // 1 - alpha
#define NBLK    (N_NODES / 16)   // 6250 16-row blocks (exact)

typedef __attribute__((ext_vector_type(2))) float v2f;
typedef __attribute__((ext_vector_type(8))) float v8f;

// ---------------- CSR build ----------------

__global__ void zero_u32(unsigned* p, int n) {
  int i = blockIdx.x * blockDim.x + threadIdx.x;
  if (i < n) p[i] = 0u;
}

__global__ void hist_rows(const int* __restrict__ rows, unsigned* __restrict__ cnt, int e) {
  int i = blockIdx.x * blockDim.x + threadIdx.x;
  if (i < e) atomicAdd(&cnt[rows[i]], 1u);
}

// Single-block sequential-chunk scan over n counts.
// In:  cnt_cursor[i] = degree of node i.
// Out: row_ptr[0..n] = exclusive prefix (CSR offsets);
//      cnt_cursor[i] overwritten with row start (scatter cursor).
__global__ void scan_counts(unsigned* __restrict__ cnt_cursor,
                            unsigned* __restrict__ row_ptr, int n) {
  __shared__ unsigned s[1024];
  __shared__ unsigned running;
  const int t = threadIdx.x;
  if (t == 0) { running = 0u; row_ptr[0] = 0u; }
  __syncthreads();
  for (int base = 0; base < n; base += 1024) {
    const int i = base + t;
    unsigned v = (i < n) ? cnt_cursor[i] : 0u;
    s[t] = v;
    __syncthreads();
    for (int off = 1; off < 1024; off <<= 1) {        // Hillis-Steele inclusive
      unsigned add = (t >= off) ? s[t - off] : 0u;
      __syncthreads();
      s[t] += add;
      __syncthreads();
    }
    const unsigned incl = s[t];
    const unsigned r = running;                        // stable until sync below
    if (i < n) {
      row_ptr[i + 1]  = r + incl;                      // exclusive prefix shifted
      cnt_cursor[i]   = r + (incl - v);                // row start offset
    }
    __syncthreads();
    if (t == 1023) running = r + s[1023];
    __syncthreads();
  }
}

// Pack (col, val-bits) per edge into its CSR slot.
__global__ void scatter_edges(const int* __restrict__ rows, const int* __restrict__ cols,
                              const float* __restrict__ vals, unsigned* __restrict__ cursor,
                              int2* __restrict__ edges, int e) {
  int i = blockIdx.x * blockDim.x + threadIdx.x;
  if (i < e) {
    unsigned pos = atomicAdd(&cursor[rows[i]], 1u);
    int2 ev;
    ev.x = cols[i];
    ev.y = __float_as_int(vals[i]);
    edges[pos] = ev;
  }
}

// ---------------- SpMM: agg[n][:] = sum_e w_e * l[col_e][:] ----------------
// One wave32 per dest node; lanes 0..24 own classes {2*lane, 2*lane+1}.
// Edge loop is wave-uniform; l-row gathers are coalesced 200B float2 reads
// that hit the 192MB L2 (whole logits matrix is 20MB).

__global__ __launch_bounds__(256) void spmm_csr(const float* __restrict__ l,
                                                const int2* __restrict__ edges,
                                                const unsigned* __restrict__ row_ptr,
                                                float* __restrict__ agg) {
  const int node = blockIdx.x * 8 + (threadIdx.x >> 5);
  const int lane = threadIdx.x & 31;
  if (node >= N_NODES) return;
  const unsigned s0 = row_ptr[node];
  const unsigned s1 = row_ptr[node + 1];
  const int c2 = lane * 2;
  const bool act = (c2 < NC);                 // lanes 0..24 carry data
  float a0 = 0.f, a1 = 0.f;
  for (unsigned k = s0; k < s1; ++k) {
    int2 ev = edges[k];                       // wave-uniform address
    float w = __int_as_float(ev.y);
    if (act) {
      const float2 lv = *(const float2*)(l + (size_t)ev.x * NC + c2); // 8B aligned
      a0 = fmaf(w, lv.x, a0);
      a1 = fmaf(w, lv.y, a1);
    }
  }
  if (act) {
    float2 o; o.x = a0; o.y = a1;
    *(float2*)(agg + (size_t)node * NC + c2) = o;
  }
}

// ---------------- WMMA combine: out = alpha*init + (1-alpha)*agg ----------------
// One wave per 16-row block; four 16x16 column tiles (n0 = 0,16,32,48).
// D = W*T + C with W = (1-alpha)*I16, split into four K=4 chunks on
// V_WMMA_F32_16X16X4_F32. Layouts per CDNA5 ISA 7.12.2:
//   C/D: elem(M = v + 8*half, N = lm) in VGPR v
//   A  : elem(M = lm, K = r + 2*half) in VGPR r
//   B  : elem(K = r + 2*half, N = lm) in VGPR r
// Columns >= 50 are guarded (ternary loads / predicated stores); the tile-guard
// branch is wave-uniform so EXEC is all-1s at every WMMA. Safe to run in place
// (outp == agg): each 16x16 region is read and written by exactly one wave.

__global__ __launch_bounds__(256) void combine_wmma(const float* __restrict__ init,
                                                    const float* __restrict__ agg,
                                                    float* __restrict__ outp) {
  const int wb = blockIdx.x * 8 + (threadIdx.x >> 5);
  if (wb >= NBLK) return;                       // wave-uniform guard
  const int lane = threadIdx.x & 31;
  const int half = lane >> 4;                   // 0 or 1
  const int lm   = lane & 15;
  const int m0   = wb * 16;

  for (int n0 = 0; n0 < 64; n0 += 16) {
    const int n  = n0 + lm;
    const bool ok = (n < NC);

    v8f acc;
#pragma unroll
    for (int v = 0; v < 8; ++v) {
      const int m = m0 + v + 8 * half;
      acc[v] = ok ? ALPHA * init[(size_t)m * NC + n] : 0.0f;
    }

#pragma unroll
    for (int j = 0; j < 4; ++j) {
      const int k0 = 4 * j + 2 * half;          // global K of this lane's A/B slots
      v2f a, b;
      a.x = (lm == k0)     ? OMA : 0.0f;        // diagonal (1-alpha)*I chunk
      a.y = (lm == k0 + 1) ? OMA : 0.0f;
      b.x = ok ? agg[(size_t)(m0 + k0)     * NC + n] : 0.0f;
      b.y = ok ? agg[(size_t)(m0 + k0 + 1) * NC + n] : 0.0f;
      acc = __builtin_amdgcn_wmma_f32_16x16x4_f32(false, a, false, b,
                                                  (short)0, acc, false, false);
    }

#pragma unroll
    for (int v = 0; v < 8; ++v) {
      const int m = m0 + v + 8 * half;
      if (ok) outp[(size_t)m * NC + n] = acc[v];
    }
  }
}

// ---------------- driver ----------------

extern "C" void kernel_launch(void* const* d_in, const int* in_sizes, int n_in,
                              void* d_out, int out_size, void* d_ws, size_t ws_size,
                              hipStream_t stream) {
  const float* logits = (const float*)d_in[0];
  const int*   erows  = (const int*)d_in[1];
  const int*   ecols  = (const int*)d_in[2];
  const float* evals  = (const float*)d_in[3];
  float*       out    = (float*)d_out;

  char* ws = (char*)d_ws;
  size_t off = 0;
  auto take = [&](size_t bytes) -> char* {
    off = (off + 255) & ~(size_t)255;
    char* p = ws + off;
    off += bytes;
    return p;
  };
  unsigned* row_ptr = (unsigned*)take((size_t)(N_NODES + 1) * 4);
  unsigned* cursor  = (unsigned*)take((size_t)N_NODES * 4);
  int2*     edges   = (int2*)    take((size_t)N_EDGES * 8);
  float*    bufA    = (float*)   take((size_t)N_NODES * NC * 4);
  float*    bufB    = (float*)   take((size_t)N_NODES * NC * 4);
  (void)ws_size; (void)in_sizes; (void)n_in; (void)out_size;

  // CSR build (rebuilt every call; graph-capture safe, all on `stream`)
  zero_u32<<<(N_NODES + 255) / 256, 256, 0, stream>>>(cursor, N_NODES);
  hist_rows<<<(N_EDGES + 255) / 256, 256, 0, stream>>>(erows, cursor, N_EDGES);
  scan_counts<<<1, 1024, 0, stream>>>(cursor, row_ptr, N_NODES);
  scatter_edges<<<(N_EDGES + 255) / 256, 256, 0, stream>>>(erows, ecols, evals,
                                                           cursor, edges, N_EDGES);

  // 5 propagation steps
  const float* l = logits;
  for (int it = 0; it < 5; ++it) {
    float* agg = (it & 1) ? bufB : bufA;
    spmm_csr<<<N_NODES / 8, 256, 0, stream>>>(l, edges, row_ptr, agg);
    float* dst = (it == 4) ? out : agg;        // last combine writes d_out
    combine_wmma<<<(NBLK + 7) / 8, 256, 0, stream>>>(logits, agg, dst);
    l = dst;
  }
}